// ConvAggregator_54580444398292
// MI455X (gfx1250) — compile-verified
//
#include <hip/hip_runtime.h>
#include <hip/hip_bf16.h>

// MI455X / gfx1250, wave32. All matrix work uses V_WMMA_F32_16X16X4_F32.
// f32 WMMA is mandatory here: the K=16 complex product reaches ~1e15, far
// beyond f16 range, and inputs/outputs are f32.

typedef float v2f __attribute__((ext_vector_type(2)));
typedef float v8f __attribute__((ext_vector_type(8)));

#define IN_DIM 128
#define HID    256
#define OUT_DIM 128
#define KNBR   16

__device__ __forceinline__ v8f wmma4(v2f a, v2f b, v8f c) {
    // D(16x16,f32) = A(16x4,f32) * B(4x16,f32) + C
    return __builtin_amdgcn_wmma_f32_16x16x4_f32(
        /*neg_a=*/false, a, /*neg_b=*/false, b,
        /*c_mod=*/(short)0, c, /*reuse_a=*/false, /*reuse_b=*/false);
}

// C/D layout: VGPR r -> (row = r + 8*(lane>>4), col = lane&15)
__device__ __forceinline__ void store_c16(float* M, v8f c, int lane) {
    const int half = lane >> 4, idx = lane & 15;
#pragma unroll
    for (int r = 0; r < 8; ++r) M[(r + 8 * half) * 16 + idx] = c[r];
}

// cos/sin(2*pi*t/16) tables; DFT:  D[j][k] = cT - i*sT,  IDFT kernel: E = cT + i*sT
__device__ const float COS16_dev[16] = {
    1.0f,  0.9238795325112867f,  0.7071067811865476f,  0.3826834323650898f,
    0.0f, -0.3826834323650898f, -0.7071067811865476f, -0.9238795325112867f,
   -1.0f, -0.9238795325112867f, -0.7071067811865476f, -0.3826834323650898f,
    0.0f,  0.3826834323650898f,  0.7071067811865476f,  0.9238795325112867f};
__device__ const float SIN16_dev[16] = {
    0.0f,  0.3826834323650898f,  0.7071067811865476f,  0.9238795325112867f,
    1.0f,  0.9238795325112867f,  0.7071067811865476f,  0.3826834323650898f,
    0.0f, -0.3826834323650898f, -0.7071067811865476f, -0.9238795325112867f,
   -1.0f, -0.9238795325112867f, -0.7071067811865476f, -0.3826834323650898f};

// ---------------------------------------------------------------------------
// Kernel A: pre = feature @ W_aff + b_aff (tile GEMM via WMMA), then per-node
// 2D FFT of the 16x16 map:  F = D * X * D  (D symmetric DFT matrix).
// 16 waves/block, 16 nodes/block. Writes Fr/Fi [N,256] to workspace.
// ---------------------------------------------------------------------------
__global__ __launch_bounds__(512) void aff_fft_kernel(
    const float* __restrict__ feature, const float* __restrict__ W_aff,
    const float* __restrict__ b_aff, float* __restrict__ Fr,
    float* __restrict__ Fi) {
    __shared__ float sPre[16][HID];     // 16 nodes x 256 (each row = 16x16 map)
    __shared__ float sY[16][2][256];    // per-wave Yr/Yi staging
    __shared__ float cT[16], sT[16];

    const int t = threadIdx.x;
    const int w = t >> 5, lane = t & 31;
    const int half = lane >> 4, idx = lane & 15, kb = 2 * half;
    const int m0 = blockIdx.x * 16;

    if (t < 16) { cT[t] = COS16_dev[t]; sT[t] = SIN16_dev[t]; }
    __syncthreads();

    // ---- Stage 1: pre tile, cols [16w, 16w+16), rows = block's 16 nodes ----
    {
        const int n0 = w * 16;
        v8f acc = {};
        const float* arow = feature + (size_t)(m0 + idx) * IN_DIM;
#pragma unroll 4
        for (int k0 = 0; k0 < IN_DIM; k0 += 4) {
            v2f a, b;
            a.x = arow[k0 + kb];
            a.y = arow[k0 + kb + 1];
            b.x = W_aff[(size_t)(k0 + kb) * HID + n0 + idx];
            b.y = W_aff[(size_t)(k0 + kb + 1) * HID + n0 + idx];
            acc = wmma4(a, b, acc);
        }
        const float bias = b_aff[n0 + idx];
#pragma unroll
        for (int r = 0; r < 8; ++r)
            sPre[r + 8 * half][n0 + idx] = acc[r] + bias;
    }
    __syncthreads();

    // ---- Stage 2: wave w FFTs node (m0+w)'s 16x16 map ----
    const float* X = sPre[w];
    float* Yr = sY[w][0];
    float* Yi = sY[w][1];

    // Y = X @ D   (Dr = cT[(j*k)&15], Di = -sT[(j*k)&15])
    v8f yr = {}, yi = {};
#pragma unroll
    for (int k0 = 0; k0 < 16; k0 += 4) {
        const int ka = k0 + kb, kc = ka + 1;
        v2f a, br, bi;
        a.x = X[idx * 16 + ka];
        a.y = X[idx * 16 + kc];
        br.x = cT[(ka * idx) & 15];  br.y = cT[(kc * idx) & 15];
        bi.x = -sT[(ka * idx) & 15]; bi.y = -sT[(kc * idx) & 15];
        yr = wmma4(a, br, yr);
        yi = wmma4(a, bi, yi);
    }
    store_c16(Yr, yr, lane);
    store_c16(Yi, yi, lane);

    // F = D @ Y :  Fr = Dr@Yr - Di@Yi,  Fi = Dr@Yi + Di@Yr
    v8f fr = {}, fi = {};
#pragma unroll
    for (int k0 = 0; k0 < 16; k0 += 4) {
        const int ka = k0 + kb, kc = ka + 1;
        v2f aR, aI, aIn, bR, bI;
        aR.x = cT[(idx * ka) & 15];  aR.y = cT[(idx * kc) & 15];   // Dr[idx][k]
        aI.x = -sT[(idx * ka) & 15]; aI.y = -sT[(idx * kc) & 15];  // Di[idx][k]
        aIn.x = -aI.x; aIn.y = -aI.y;                              // -Di
        bR.x = Yr[ka * 16 + idx]; bR.y = Yr[kc * 16 + idx];
        bI.x = Yi[ka * 16 + idx]; bI.y = Yi[kc * 16 + idx];
        fr = wmma4(aR, bR, fr);
        fr = wmma4(aIn, bI, fr);
        fi = wmma4(aR, bI, fi);
        fi = wmma4(aI, bR, fi);
    }
    float* frg = Fr + (size_t)(m0 + w) * 256;
    float* fig = Fi + (size_t)(m0 + w) * 256;
#pragma unroll
    for (int r = 0; r < 8; ++r) {
        frg[(r + 8 * half) * 16 + idx] = fr[r];
        fig[(r + 8 * half) * 16 + idx] = fi[r];
    }
}

// ---------------------------------------------------------------------------
// Kernel B: gather F of 16 neighbors, elementwise complex product, inverse
// 2D FFT (h = real(E*P*E)/256), then out = h @ W_mlp + b_mlp via WMMA.
// 16 waves/block, 16 destination nodes/block.
// ---------------------------------------------------------------------------
__global__ __launch_bounds__(512) void prod_ifft_mlp_kernel(
    const float* __restrict__ Fr, const float* __restrict__ Fi,
    const int* __restrict__ nbr, const float* __restrict__ W_mlp,
    const float* __restrict__ b_mlp, float* __restrict__ out) {
    __shared__ float sS[16][2][256];    // per-wave {Pr,Pi} -> {Tr,Ti} -> h
    __shared__ float cT[16], sT[16];

    const int t = threadIdx.x;
    const int w = t >> 5, lane = t & 31;
    const int half = lane >> 4, idx = lane & 15, kb = 2 * half;
    const int m0 = blockIdx.x * 16;
    const int node = m0 + w;

    if (t < 16) { cT[t] = COS16_dev[t]; sT[t] = SIN16_dev[t]; }
    __syncthreads();

    // ---- complex product over K neighbors (per-lane, 8 elems each) ----
    float pr[8], pi[8];
#pragma unroll
    for (int j = 0; j < 8; ++j) { pr[j] = 1.0f; pi[j] = 0.0f; }
    for (int kk = 0; kk < KNBR; ++kk) {
        const int s = nbr[(size_t)node * KNBR + kk];
        const float* fr = Fr + (size_t)s * 256;
        const float* fi = Fi + (size_t)s * 256;
#pragma unroll
        for (int j = 0; j < 8; ++j) {
            const float ar = fr[j * 32 + lane], ai = fi[j * 32 + lane];
            const float nr = pr[j] * ar - pi[j] * ai;
            const float ni = pr[j] * ai + pi[j] * ar;
            pr[j] = nr; pi[j] = ni;
        }
    }
    float* Pr = sS[w][0];
    float* Pi = sS[w][1];
#pragma unroll
    for (int j = 0; j < 8; ++j) {
        Pr[j * 32 + lane] = pr[j];
        Pi[j * 32 + lane] = pi[j];
    }

    // ---- T = E @ P  (Er = cT, Ei = +sT) ----
    v8f tr = {}, ti = {};
#pragma unroll
    for (int k0 = 0; k0 < 16; k0 += 4) {
        const int ka = k0 + kb, kc = ka + 1;
        v2f aR, aI, aIn, bR, bI;
        aR.x = cT[(idx * ka) & 15]; aR.y = cT[(idx * kc) & 15];
        aI.x = sT[(idx * ka) & 15]; aI.y = sT[(idx * kc) & 15];
        aIn.x = -aI.x; aIn.y = -aI.y;
        bR.x = Pr[ka * 16 + idx]; bR.y = Pr[kc * 16 + idx];
        bI.x = Pi[ka * 16 + idx]; bI.y = Pi[kc * 16 + idx];
        tr = wmma4(aR, bR, tr);
        tr = wmma4(aIn, bI, tr);   // -Ei @ Pi
        ti = wmma4(aR, bI, ti);
        ti = wmma4(aI, bR, ti);
    }
    store_c16(Pr, tr, lane);       // overwrite P with T (all P reads done)
    store_c16(Pi, ti, lane);

    // ---- h = real(T @ E) = Tr@Er - Ti@Ei ----
    v8f h = {};
#pragma unroll
    for (int k0 = 0; k0 < 16; k0 += 4) {
        const int ka = k0 + kb, kc = ka + 1;
        v2f aTr, aTi, bR, bIn;
        aTr.x = Pr[idx * 16 + ka]; aTr.y = Pr[idx * 16 + kc];
        aTi.x = Pi[idx * 16 + ka]; aTi.y = Pi[idx * 16 + kc];
        bR.x = cT[(ka * idx) & 15];   bR.y = cT[(kc * idx) & 15];
        bIn.x = -sT[(ka * idx) & 15]; bIn.y = -sT[(kc * idx) & 15]; // -Ei
        h = wmma4(aTr, bR, h);
        h = wmma4(aTi, bIn, h);
    }
#pragma unroll
    for (int r = 0; r < 8; ++r)                  // h row into sS[w][0]
        Pr[(r + 8 * half) * 16 + idx] = h[r] * (1.0f / 256.0f);
    __syncthreads();

    // ---- out tile [16 nodes x 128] = h @ W_mlp + b (waves 0..7) ----
    if (w < 8) {
        const int n0 = w * 16;
        v8f acc = {};
#pragma unroll 4
        for (int k0 = 0; k0 < HID; k0 += 4) {
            v2f a, b;
            a.x = sS[idx][0][k0 + kb];
            a.y = sS[idx][0][k0 + kb + 1];
            b.x = W_mlp[(size_t)(k0 + kb) * OUT_DIM + n0 + idx];
            b.y = W_mlp[(size_t)(k0 + kb + 1) * OUT_DIM + n0 + idx];
            acc = wmma4(a, b, acc);
        }
        const float bias = b_mlp[n0 + idx];
#pragma unroll
        for (int r = 0; r < 8; ++r)
            out[(size_t)(m0 + r + 8 * half) * OUT_DIM + n0 + idx] = acc[r] + bias;
    }
}

extern "C" void kernel_launch(void* const* d_in, const int* in_sizes, int n_in,
                              void* d_out, int out_size, void* d_ws, size_t ws_size,
                              hipStream_t stream) {
    const float* feature = (const float*)d_in[0];
    const int*   nbr     = (const int*)d_in[1];
    const float* W_aff   = (const float*)d_in[2];
    const float* b_aff   = (const float*)d_in[3];
    const float* W_mlp   = (const float*)d_in[4];
    const float* b_mlp   = (const float*)d_in[5];
    float* out = (float*)d_out;

    const int N = in_sizes[0] / IN_DIM;     // 20000 (divisible by 16)
    const int blocks = N / 16;              // 1250

    float* wsf = (float*)d_ws;
    float* Fr = wsf;                        // [N,256]
    float* Fi = wsf + (size_t)N * 256;      // [N,256]

    aff_fft_kernel<<<blocks, 512, 0, stream>>>(feature, W_aff, b_aff, Fr, Fi);
    prod_ifft_mlp_kernel<<<blocks, 512, 0, stream>>>(Fr, Fi, nbr, W_mlp, b_mlp, out);
}